// GAT_73658689126820
// MI455X (gfx1250) — compile-verified
//
#include <hip/hip_runtime.h>
#include <hip/hip_bf16.h>
#include <math.h>

typedef __attribute__((ext_vector_type(2))) float v2f;
typedef __attribute__((ext_vector_type(8))) float v8f;

#define NEG_SLOPE 0.2f

// Monotonic float -> unsigned key so segment-max can use global_atomic_max_u32.
// key 0u decodes below every real float, so 0-init is "-inf".
__device__ __forceinline__ unsigned fkey(float f) {
  unsigned u = __float_as_uint(f);
  return (u & 0x80000000u) ? ~u : (u | 0x80000000u);
}
__device__ __forceinline__ float fdecode(unsigned k) {
  return (k & 0x80000000u) ? __uint_as_float(k & 0x7fffffffu)
                           : __uint_as_float(~k);
}

// ---------------------------------------------------------------------------
// Dense GEMM C[M,NCOLS] = A[M,K] * B[K,LDB] via chained v_wmma_f32_16x16x4_f32,
// one wave32 per 16x16 tile. LDB/NCOLS are compile-time so all loop loads and
// stores use immediate offsets. B must be zero-padded to LDB >= ntiles_n*16.
// A-frag (16x4 fp32): lanes 0-15 -> K={kb,kb+1}; lanes 16-31 -> K={kb+2,kb+3}.
// C/D 16x16: vgpr i holds M=i (lanes 0-15) and M=i+8 (lanes 16-31).
// ---------------------------------------------------------------------------
template <int K, int LDB, int NCOLS>
__global__ void gat_gemm_wmma(const float* __restrict__ A,
                              const float* __restrict__ B,
                              float* __restrict__ C,
                              int M, int ntiles_n) {
  int wave = (int)((blockIdx.x * (unsigned)blockDim.x + threadIdx.x) >> 5);
  int lane = threadIdx.x & 31;
  int mtiles = (M + 15) >> 4;
  if (wave >= mtiles * ntiles_n) return;
  int mt = wave / ntiles_n;
  int nt = wave - mt * ntiles_n;
  int m0 = mt << 4;
  int n0 = nt << 4;
  int half = lane >> 4;   // 0: K pair {kb,kb+1}; 1: K pair {kb+2,kb+3}
  int l = lane & 15;

  int arow = m0 + l;
  if (arow >= M) arow = M - 1;               // clamp; stores are guarded
  const float* Ap = A + (size_t)arow * K + (half << 1);
  int col = n0 + l;
  const float* Bp = B + (size_t)(half << 1) * LDB + col;

  v8f c = {};
#pragma unroll
  for (int kb = 0; kb < K; kb += 4) {
    v2f a = *(const v2f*)(Ap + kb);          // global_load_b64, imm offset
    v2f b;
    b.x = Bp[kb * LDB];                      // imm offsets
    b.y = Bp[(kb + 1) * LDB];
    c = __builtin_amdgcn_wmma_f32_16x16x4_f32(false, a, false, b, (short)0, c,
                                              false, false);
  }

  bool colok = (NCOLS >= 16) ? true : (l < NCOLS);
  if (colok) {
    float* Cp = C + (size_t)(m0 + (half << 3)) * NCOLS + col;
    if (m0 + 16 <= M) {                      // full tile: unconditional stores
#pragma unroll
      for (int i = 0; i < 8; ++i) Cp[i * NCOLS] = c[i];
    } else {
#pragma unroll
      for (int i = 0; i < 8; ++i)
        if (m0 + i + (half << 3) < M) Cp[i * NCOLS] = c[i];
    }
  }
}

// Zero-pad W2 [64,2] into W2p [64,16].
__global__ void gat_pad_w2(const float* __restrict__ W2,
                           float* __restrict__ W2p) {
  int i = blockIdx.x * blockDim.x + threadIdx.x;
  if (i >= 64 * 16) return;
  int k = i >> 4;
  int cidx = i & 15;
  W2p[i] = (cidx < 2) ? W2[k * 2 + cidx] : 0.f;
}

// ---------------------------------------------------------------------------
// Layer 1 (H=8, C=8): per-(node,head) attention coefficients + state init.
// ---------------------------------------------------------------------------
__global__ void gat_scores1(const float* __restrict__ h1,
                            const float* __restrict__ att_src,
                            const float* __restrict__ att_dst,
                            float* __restrict__ as, float* __restrict__ ad,
                            unsigned* __restrict__ key,
                            float* __restrict__ sum, float* __restrict__ acc,
                            int N) {
  int i = blockIdx.x * blockDim.x + threadIdx.x;
  if (i >= N * 8) return;
  int n = i >> 3;
  int h = i & 7;
  float hv[8];
  *(float4*)&hv[0] = *(const float4*)(h1 + (size_t)n * 64 + h * 8);
  *(float4*)&hv[4] = *(const float4*)(h1 + (size_t)n * 64 + h * 8 + 4);
  float s = 0.f, d = 0.f;
#pragma unroll
  for (int c = 0; c < 8; ++c) {
    s += hv[c] * att_src[h * 8 + c];
    d += hv[c] * att_dst[h * 8 + c];
  }
  as[i] = s;
  ad[i] = d;
  key[i] = 0u;
  sum[i] = 0.f;
  float4 z = {0.f, 0.f, 0.f, 0.f};
  *(float4*)(acc + (size_t)n * 64 + h * 8) = z;
  *(float4*)(acc + (size_t)n * 64 + h * 8 + 4) = z;
}

// Edge pass A (layer 1): segment max over destinations; one thread per edge,
// vector (b128) score loads, 8 atomic u32-max per edge.
__global__ void gat_edge_max1(const long long* __restrict__ src,
                              const long long* __restrict__ dst,
                              const float* __restrict__ as,
                              const float* __restrict__ ad,
                              unsigned* __restrict__ key, int E, int N) {
  int e = blockIdx.x * blockDim.x + threadIdx.x;
  if (e >= E + N) return;
  int s, d;
  if (e < E) { s = (int)src[e]; d = (int)dst[e]; }
  else       { s = d = e - E; }
  float av[8], dv[8];
  *(float4*)&av[0] = *(const float4*)(as + (size_t)s * 8);
  *(float4*)&av[4] = *(const float4*)(as + (size_t)s * 8 + 4);
  *(float4*)&dv[0] = *(const float4*)(ad + (size_t)d * 8);
  *(float4*)&dv[4] = *(const float4*)(ad + (size_t)d * 8 + 4);
  unsigned* kp = key + (size_t)d * 8;
#pragma unroll
  for (int h = 0; h < 8; ++h) {
    float v = av[h] + dv[h];
    v = v > 0.f ? v : NEG_SLOPE * v;
    atomicMax(&kp[h], fkey(v));
  }
}

// Decode unsigned max-keys back to floats in place.
__global__ void gat_decode(unsigned* __restrict__ key, int n) {
  int i = blockIdx.x * blockDim.x + threadIdx.x;
  if (i >= n) return;
  float m = fdecode(key[i]);
  ((float*)key)[i] = m;
}

// Edge pass B (layer 1): accumulate exp-weights and unnormalized messages.
__global__ void gat_edge_sum1(const long long* __restrict__ src,
                              const long long* __restrict__ dst,
                              const float* __restrict__ as,
                              const float* __restrict__ ad,
                              const float* __restrict__ maxf,
                              const float* __restrict__ h1,
                              float* __restrict__ sum,
                              float* __restrict__ acc, int E, int N) {
  int e = blockIdx.x * blockDim.x + threadIdx.x;
  if (e >= E + N) return;
  int s, d;
  if (e < E) { s = (int)src[e]; d = (int)dst[e]; }
  else       { s = d = e - E; }
  float av[8], dv[8], mv[8];
  *(float4*)&av[0] = *(const float4*)(as + (size_t)s * 8);
  *(float4*)&av[4] = *(const float4*)(as + (size_t)s * 8 + 4);
  *(float4*)&dv[0] = *(const float4*)(ad + (size_t)d * 8);
  *(float4*)&dv[4] = *(const float4*)(ad + (size_t)d * 8 + 4);
  *(float4*)&mv[0] = *(const float4*)(maxf + (size_t)d * 8);
  *(float4*)&mv[4] = *(const float4*)(maxf + (size_t)d * 8 + 4);
  const float* hbase = h1 + (size_t)s * 64;
  float* abase = acc + (size_t)d * 64;
  float* sp = sum + (size_t)d * 8;
#pragma unroll
  for (int h = 0; h < 8; ++h) {
    float v = av[h] + dv[h];
    v = v > 0.f ? v : NEG_SLOPE * v;
    float ex = expf(v - mv[h]);
    atomicAdd(&sp[h], ex);
    float hv[8];
    *(float4*)&hv[0] = *(const float4*)(hbase + h * 8);
    *(float4*)&hv[4] = *(const float4*)(hbase + h * 8 + 4);
    float* ap = abase + h * 8;
#pragma unroll
    for (int c = 0; c < 8; ++c) atomicAdd(&ap[c], ex * hv[c]);
  }
}

// Normalize, add bias, ELU -> input features of layer 2 (in place in acc).
__global__ void gat_fin1(float* __restrict__ acc, const float* __restrict__ sum,
                         const float* __restrict__ b1, int N) {
  int i = blockIdx.x * blockDim.x + threadIdx.x;
  if (i >= N * 64) return;
  int n = i >> 6;
  int hc = i & 63;
  int h = hc >> 3;
  float v = acc[i] / sum[n * 8 + h] + b1[hc];
  acc[i] = v > 0.f ? v : expm1f(v);
}

// ---------------------------------------------------------------------------
// Layer 2 (H=1, C=2)
// ---------------------------------------------------------------------------
__global__ void gat_scores2(const float* __restrict__ h2,
                            const float* __restrict__ att_src,
                            const float* __restrict__ att_dst,
                            float* __restrict__ as, float* __restrict__ ad,
                            unsigned* __restrict__ key,
                            float* __restrict__ sum, float* __restrict__ acc,
                            int N) {
  int n = blockIdx.x * blockDim.x + threadIdx.x;
  if (n >= N) return;
  float v0 = h2[(size_t)n * 2 + 0];
  float v1 = h2[(size_t)n * 2 + 1];
  as[n] = v0 * att_src[0] + v1 * att_src[1];
  ad[n] = v0 * att_dst[0] + v1 * att_dst[1];
  key[n] = 0u;
  sum[n] = 0.f;
  acc[(size_t)n * 2 + 0] = 0.f;
  acc[(size_t)n * 2 + 1] = 0.f;
}

__global__ void gat_edge_max2(const long long* __restrict__ src,
                              const long long* __restrict__ dst,
                              const float* __restrict__ as,
                              const float* __restrict__ ad,
                              unsigned* __restrict__ key, int E, int N) {
  int e = blockIdx.x * blockDim.x + threadIdx.x;
  if (e >= E + N) return;
  int s, d;
  if (e < E) { s = (int)src[e]; d = (int)dst[e]; }
  else       { s = d = e - E; }
  float v = as[s] + ad[d];
  v = v > 0.f ? v : NEG_SLOPE * v;
  atomicMax(&key[d], fkey(v));
}

__global__ void gat_edge_sum2(const long long* __restrict__ src,
                              const long long* __restrict__ dst,
                              const float* __restrict__ as,
                              const float* __restrict__ ad,
                              const float* __restrict__ maxf,
                              const float* __restrict__ h2,
                              float* __restrict__ sum,
                              float* __restrict__ acc, int E, int N) {
  int e = blockIdx.x * blockDim.x + threadIdx.x;
  if (e >= E + N) return;
  int s, d;
  if (e < E) { s = (int)src[e]; d = (int)dst[e]; }
  else       { s = d = e - E; }
  float v = as[s] + ad[d];
  v = v > 0.f ? v : NEG_SLOPE * v;
  float ex = expf(v - maxf[d]);
  atomicAdd(&sum[d], ex);
  atomicAdd(&acc[(size_t)d * 2 + 0], ex * h2[(size_t)s * 2 + 0]);
  atomicAdd(&acc[(size_t)d * 2 + 1], ex * h2[(size_t)s * 2 + 1]);
}

__global__ void gat_fin2(const float* __restrict__ acc,
                         const float* __restrict__ sum,
                         const float* __restrict__ b2,
                         float* __restrict__ out, int N) {
  int n = blockIdx.x * blockDim.x + threadIdx.x;
  if (n >= N) return;
  float inv = 1.f / sum[n];
  float v0 = acc[(size_t)n * 2 + 0] * inv + b2[0];
  float v1 = acc[(size_t)n * 2 + 1] * inv + b2[1];
  float m = fmaxf(v0, v1);
  float lse = m + logf(expf(v0 - m) + expf(v1 - m));
  out[(size_t)n * 2 + 0] = v0 - lse;
  out[(size_t)n * 2 + 1] = v1 - lse;
}

// ---------------------------------------------------------------------------
static inline unsigned cdiv(long long a, int b) {
  return (unsigned)((a + b - 1) / b);
}

extern "C" void kernel_launch(void* const* d_in, const int* in_sizes, int n_in,
                              void* d_out, int out_size, void* d_ws,
                              size_t ws_size, hipStream_t stream) {
  const float* x        = (const float*)d_in[0];
  const long long* ei   = (const long long*)d_in[1];
  const float* W1       = (const float*)d_in[2];
  const float* att_src1 = (const float*)d_in[3];
  const float* att_dst1 = (const float*)d_in[4];
  const float* b1       = (const float*)d_in[5];
  const float* W2       = (const float*)d_in[6];
  const float* att_src2 = (const float*)d_in[7];
  const float* att_dst2 = (const float*)d_in[8];
  const float* b2       = (const float*)d_in[9];
  float* out = (float*)d_out;

  const int N = in_sizes[0] / 32;
  const int E = in_sizes[1] / 2;
  const long long ET = (long long)E + N;

  // Workspace carve (floats), ~168 floats per node + 1K pad for W2.
  float* h1   = (float*)d_ws;                 // N*64
  float* as1  = h1 + (size_t)N * 64;          // N*8
  float* ad1  = as1 + (size_t)N * 8;          // N*8
  float* sum1 = ad1 + (size_t)N * 8;          // N*8
  float* acc1 = sum1 + (size_t)N * 8;         // N*64 (becomes layer-2 input)
  float* h2   = acc1 + (size_t)N * 64;        // N*2
  float* as2  = h2 + (size_t)N * 2;           // N
  float* ad2  = as2 + (size_t)N;              // N
  float* sum2 = ad2 + (size_t)N;              // N
  float* acc2 = sum2 + (size_t)N;             // N*2
  unsigned* key1 = (unsigned*)(acc2 + (size_t)N * 2);  // N*8
  unsigned* key2 = key1 + (size_t)N * 8;               // N
  float* W2p  = (float*)(key2 + (size_t)N);   // 64*16

  const int BLK = 256;
  const int mtiles = (N + 15) >> 4;

  // 0) pad W2 to 16 columns so GEMM2 B-loads need no bounds checks
  gat_pad_w2<<<4, BLK, 0, stream>>>(W2, W2p);
  // 1) h1 = x @ W1  (K=32, 64 cols, 4 N-tiles) via fp32 WMMA
  gat_gemm_wmma<32, 64, 64>
      <<<cdiv((long long)mtiles * 4 * 32, BLK), BLK, 0, stream>>>(
          x, W1, h1, N, 4);
  // 2) attention coefficients per (node,head) + init max/sum/acc
  gat_scores1<<<cdiv((long long)N * 8, BLK), BLK, 0, stream>>>(
      h1, att_src1, att_dst1, as1, ad1, key1, sum1, acc1, N);
  // 3) segment max over incoming edges
  gat_edge_max1<<<cdiv(ET, BLK), BLK, 0, stream>>>(ei, ei + E, as1, ad1, key1,
                                                   E, N);
  // 4) decode keys -> float max
  gat_decode<<<cdiv((long long)N * 8, BLK), BLK, 0, stream>>>(key1, N * 8);
  // 5) exp-weights + unnormalized message aggregation
  gat_edge_sum1<<<cdiv(ET, BLK), BLK, 0, stream>>>(
      ei, ei + E, as1, ad1, (const float*)key1, h1, sum1, acc1, E, N);
  // 6) normalize + bias + ELU (in place -> layer-2 input features)
  gat_fin1<<<cdiv((long long)N * 64, BLK), BLK, 0, stream>>>(acc1, sum1, b1, N);
  // 7) h2 = elu(out1) @ W2p  (K=64, 2 real cols of 16-wide pad) via fp32 WMMA
  gat_gemm_wmma<64, 16, 2>
      <<<cdiv((long long)mtiles * 32, BLK), BLK, 0, stream>>>(
          acc1, W2p, h2, N, 1);
  // 8-11) layer-2 attention (H=1, C=2)
  gat_scores2<<<cdiv(N, BLK), BLK, 0, stream>>>(h2, att_src2, att_dst2, as2,
                                                ad2, key2, sum2, acc2, N);
  gat_edge_max2<<<cdiv(ET, BLK), BLK, 0, stream>>>(ei, ei + E, as2, ad2, key2,
                                                   E, N);
  gat_decode<<<cdiv(N, BLK), BLK, 0, stream>>>(key2, N);
  gat_edge_sum2<<<cdiv(ET, BLK), BLK, 0, stream>>>(
      ei, ei + E, as2, ad2, (const float*)key2, h2, sum2, acc2, E, N);
  // 12) normalize + bias + log_softmax -> d_out
  gat_fin2<<<cdiv(N, BLK), BLK, 0, stream>>>(acc2, sum2, b2, out, N);
}